// OuterPropagation_38646115729711
// MI455X (gfx1250) — compile-verified
//
#include <hip/hip_runtime.h>

#define N_NODES   100000
#define N_HEDGES  20000
#define N_EDGES   600000
#define IN_DIM    128
#define OUT_DIM   128
#define K_DIM     256          // 2 * IN_DIM
#define KSTEPS    64           // K_DIM / 4
#define NTILES    8            // OUT_DIM / 16
#define PF        8            // A-fragment prefetch depth (register ring)

typedef float v2f __attribute__((ext_vector_type(2)));
typedef float v8f __attribute__((ext_vector_type(8)));

// ---------------- Pass 1: init per-node last-edge index ----------------
__global__ void op_init_last(int* __restrict__ last) {
    int i = blockIdx.x * blockDim.x + threadIdx.x;
    if (i < N_NODES) last[i] = -1;
}

// ---------------- Pass 2: segment-max over edges (scatter) -------------
__global__ void op_scatter_last(const int* __restrict__ links, int* __restrict__ last) {
    int e = blockIdx.x * blockDim.x + threadIdx.x;
    if (e < N_EDGES) {
        int n = links[3 * e + 0];
        atomicMax(&last[n], e);
    }
}

// ---------------- Pass 3: fused gather + [f|x] @ W_f via WMMA ----------
// Block: 256 threads = 8 waves. Wave w computes rows [blk*128 + 16w, +16) x all 128 cols.
// Dynamic LDS holds W_f pre-swizzled into WMMA B fragments:
//   bfrag[(kk*8 + nt)*32 + lane] = { Wf[4kk+sub][nt*16+(lane&15)],
//                                    Wf[4kk+sub+1][nt*16+(lane&15)] },  sub = (lane>>4)*2
// A fragments stream from global through an 8-deep register prefetch ring; invalid
// rows use clamped safe pointers and are masked to zero at store time (ballot mask),
// so the K loop runs with full EXEC and zero divergence.
__global__ void op_gather_gemm(const float* __restrict__ xemb,
                               const float* __restrict__ hemb,
                               const int*   __restrict__ links,
                               const float* __restrict__ Wf,
                               const int*   __restrict__ last,
                               float*       __restrict__ out) {
    extern __shared__ float2 bfrag[];   // 64 * 8 * 32 float2 = 128 KB

    const int tid  = threadIdx.x;
    const int lane = tid & 31;
    const int wave = tid >> 5;

    // ---- stage W_f fragments into LDS (once per block) ----
    for (int i = tid; i < KSTEPS * NTILES * 32; i += 256) {
        const int l   = i & 31;
        const int nt  = (i >> 5) & 7;
        const int kk  = i >> 8;
        const int sb  = (l >> 4) << 1;            // 0 or 2
        const int col = nt * 16 + (l & 15);
        const int r0  = 4 * kk + sb;
        float2 v;
        v.x = Wf[r0 * OUT_DIM + col];
        v.y = Wf[(r0 + 1) * OUT_DIM + col];
        bfrag[i] = v;
    }
    __syncthreads();

    // ---- per-lane A-row setup (lanes L and L+16 share row L&15) ----
    const int mbase = blockIdx.x * 128 + wave * 16;
    const int row   = mbase + (lane & 15);
    const int rowc  = (row < N_NODES) ? row : (N_NODES - 1);   // clamped, always safe
    int e = last[rowc];
    const bool valid = (row < N_NODES) && (e >= 0);
    if (e < 0) e = 0;                                          // clamped, always safe
    const int selh = links[3 * e + 2];
    const unsigned vmask = __builtin_amdgcn_ballot_w32(valid); // bit i = validity of tile row i (0..15)

    const int sub = (lane >> 4) << 1;             // K sub-offset for A fragment
    const float* base_f = hemb + (size_t)selh * IN_DIM + sub;  // concat cols 0..127
    const float* base_x = xemb + (size_t)rowc * IN_DIM + sub;  // concat cols 128..255

    // ---- prefetch ring prologue: steps 0..PF-1 (all within f-part: 4*7+3 < 128) ----
    v2f abuf[PF];
    #pragma unroll
    for (int p = 0; p < PF; ++p) {
        const float* s = base_f + 4 * p;
        abuf[p].x = s[0];
        abuf[p].y = s[1];
    }

    v8f acc[NTILES];
    #pragma unroll
    for (int nt = 0; nt < NTILES; ++nt)
        acc[nt] = (v8f){0.f, 0.f, 0.f, 0.f, 0.f, 0.f, 0.f, 0.f};

    // ---- K loop: 64 steps of K=4; 8 N-tiles per step; PF-deep A pipeline ----
    for (int kb = 0; kb < KSTEPS; kb += PF) {
        #pragma unroll
        for (int p = 0; p < PF; ++p) {
            const int kk = kb + p;
            const v2f a = abuf[p];
            const int nk = kk + PF;
            if (nk < KSTEPS) {                    // uniform scalar condition
                const float* s = (nk < KSTEPS / 2) ? (base_f + 4 * nk)
                                                   : (base_x + (4 * nk - IN_DIM));
                abuf[p].x = s[0];
                abuf[p].y = s[1];
            }
            #pragma unroll
            for (int nt = 0; nt < NTILES; ++nt) {
                float2 bf = bfrag[(kk * NTILES + nt) * 32 + lane];
                v2f b;
                b.x = bf.x;
                b.y = bf.y;
                acc[nt] = __builtin_amdgcn_wmma_f32_16x16x4_f32(
                    /*neg_a=*/false, a, /*neg_b=*/false, b,
                    /*c_mod=*/(short)0, acc[nt],
                    /*reuse_a=*/false, /*reuse_b=*/false);
            }
        }
    }

    // ---- store C: VGPR v -> M = mbase + v + 8*(lane>=16), N = nt*16 + (lane&15) ----
    const int half  = (lane >> 4) << 3;           // 0 or 8
    const int rbase = mbase + half;
    const int ncol  = lane & 15;
    #pragma unroll
    for (int nt = 0; nt < NTILES; ++nt) {
        #pragma unroll
        for (int v = 0; v < 8; ++v) {
            const int r = rbase + v;
            const bool rv = (vmask >> (v + half)) & 1u;
            const float val = rv ? acc[nt][v] : 0.0f;
            if (r < N_NODES)
                out[(size_t)r * OUT_DIM + nt * 16 + ncol] = val;
        }
    }
}

extern "C" void kernel_launch(void* const* d_in, const int* in_sizes, int n_in,
                              void* d_out, int out_size, void* d_ws, size_t ws_size,
                              hipStream_t stream) {
    const float* node_emb = (const float*)d_in[0];
    // d_in[1] (semalink_embeddings) and d_in[4] (W_a) provably do not affect the
    // output: softmax over a length-1 axis is identically 1.0, so gamma == 1.
    const float* hedge_emb = (const float*)d_in[2];
    const int*   links     = (const int*)d_in[3];
    const float* Wf        = (const float*)d_in[5];
    float*       out       = (float*)d_out;

    int* last = (int*)d_ws;   // N_NODES ints of scratch

    // Pass 1: init segment-max identity
    op_init_last<<<(N_NODES + 255) / 256, 256, 0, stream>>>(last);

    // Pass 2: last[n] = max e with n_idx[e] == n
    op_scatter_last<<<(N_EDGES + 255) / 256, 256, 0, stream>>>(links, last);

    // Pass 3: fused gather + GEMM (128 rows per block, 8 waves x 16-row M-tiles)
    const int smem_bytes = KSTEPS * NTILES * 32 * (int)sizeof(float2);   // 131072
    hipFuncSetAttribute((const void*)op_gather_gemm,
                        hipFuncAttributeMaxDynamicSharedMemorySize, smem_bytes);
    const int nblocks = (N_NODES + 127) / 128;
    op_gather_gemm<<<nblocks, 256, smem_bytes, stream>>>(
        node_emb, hedge_emb, links, Wf, last, out);
}